// MAGCN_54296976556534
// MI455X (gfx1250) — compile-verified
//
#include <hip/hip_runtime.h>
#include <hip/hip_bf16.h>

typedef __attribute__((ext_vector_type(2))) float v2f;
typedef __attribute__((ext_vector_type(8))) float v8f;

#define GCN_IN_DIM 512
#define GCN_HIDDEN 128
#define GCN_OUT_DIM 64
#define GCN_EPS 0.3f

// ---------------------------------------------------------------------------
// GEMM1 + ReLU:  out[M,128] = relu(h[M,512] @ w[128,512]^T + b)
// Block = 256 threads = 8 waves; each wave owns one 16-col N-tile.
// A tile (16x512) staged in padded LDS, f32 WMMA 16x16x4 over K.
// ---------------------------------------------------------------------------
__global__ __launch_bounds__(256) void gemm1_relu_kernel(
    const float* __restrict__ h, const float* __restrict__ w,
    const float* __restrict__ b, float* __restrict__ out) {
  __shared__ float As[16 * 516];  // +4 pad -> conflict-free fragment reads
  const int m0 = blockIdx.x * 16;

  // Cooperative A-tile load: 16 rows x 512 floats as float4
  {
    const float4* src4 = (const float4*)(h + (size_t)m0 * GCN_IN_DIM);
    for (int i = threadIdx.x; i < 16 * (GCN_IN_DIM / 4); i += 256) {
      const int row = i >> 7;          // 128 float4 per row
      const int c4  = i & 127;
      ((float4*)(As + row * 516))[c4] = src4[row * 128 + c4];
    }
  }
  __syncthreads();

  const int wave  = threadIdx.x >> 5;
  const int lane  = threadIdx.x & 31;
  const int n0    = wave * 16;
  const int frag  = lane & 15;           // A: M row, B: N col
  const int khalf = (lane >> 4) * 2;     // K sub-offset per lane half

  const float* abase = As + frag * 516 + khalf;
  const float* bbase = w + (size_t)(n0 + frag) * GCN_IN_DIM + khalf;

  v8f c = {};
#pragma unroll 4
  for (int k0 = 0; k0 < GCN_IN_DIM; k0 += 4) {
    v2f a, bb;
    a.x  = abase[k0 + 0];
    a.y  = abase[k0 + 1];
    bb.x = bbase[k0 + 0];
    bb.y = bbase[k0 + 1];
    c = __builtin_amdgcn_wmma_f32_16x16x4_f32(false, a, false, bb,
                                              (short)0, c, false, false);
  }

  const float bias  = b[n0 + frag];
  const int   rbase = (lane >> 4) * 8;
#pragma unroll
  for (int r = 0; r < 8; ++r) {
    float v = c[r] + bias;
    v = v > 0.f ? v : 0.f;
    out[(size_t)(m0 + rbase + r) * GCN_HIDDEN + n0 + frag] = v;
  }
}

// ---------------------------------------------------------------------------
// GEMM2:  out[M,64] = h[M,128] @ w[64,128]^T + b   (logits, no activation)
// Block = 128 threads = 4 waves (N = 4 tiles of 16).
// ---------------------------------------------------------------------------
__global__ __launch_bounds__(128) void gemm2_kernel(
    const float* __restrict__ h, const float* __restrict__ w,
    const float* __restrict__ b, float* __restrict__ out) {
  __shared__ float As[16 * 132];
  const int m0 = blockIdx.x * 16;
  {
    const float4* src4 = (const float4*)(h + (size_t)m0 * GCN_HIDDEN);
    for (int i = threadIdx.x; i < 16 * (GCN_HIDDEN / 4); i += 128) {
      const int row = i >> 5;          // 32 float4 per row
      const int c4  = i & 31;
      ((float4*)(As + row * 132))[c4] = src4[row * 32 + c4];
    }
  }
  __syncthreads();

  const int wave  = threadIdx.x >> 5;
  const int lane  = threadIdx.x & 31;
  const int n0    = wave * 16;
  const int frag  = lane & 15;
  const int khalf = (lane >> 4) * 2;

  const float* abase = As + frag * 132 + khalf;
  const float* bbase = w + (size_t)(n0 + frag) * GCN_HIDDEN + khalf;

  v8f c = {};
#pragma unroll 4
  for (int k0 = 0; k0 < GCN_HIDDEN; k0 += 4) {
    v2f a, bb;
    a.x  = abase[k0 + 0];
    a.y  = abase[k0 + 1];
    bb.x = bbase[k0 + 0];
    bb.y = bbase[k0 + 1];
    c = __builtin_amdgcn_wmma_f32_16x16x4_f32(false, a, false, bb,
                                              (short)0, c, false, false);
  }

  const float bias  = b[n0 + frag];
  const int   rbase = (lane >> 4) * 8;
#pragma unroll
  for (int r = 0; r < 8; ++r) {
    out[(size_t)(m0 + rbase + r) * GCN_OUT_DIM + n0 + frag] = c[r] + bias;
  }
}

// ---------------------------------------------------------------------------
// Gate scores: s_dst[n] = h[n,:]·w_dst, s_src[n] = h[n,:]·w_src
// One wave per node; lane handles 4 contiguous channels; shfl_xor reduce.
// ---------------------------------------------------------------------------
__global__ __launch_bounds__(256) void gate_scores_kernel(
    const float* __restrict__ h, const float* __restrict__ gw,
    float* __restrict__ sdst, float* __restrict__ ssrc, int n) {
  const int node = (int)((blockIdx.x * (unsigned)blockDim.x + threadIdx.x) >> 5);
  const int lane = threadIdx.x & 31;
  if (node >= n) return;

  const float4 a  = ((const float4*)(h + (size_t)node * GCN_HIDDEN))[lane];
  const float4 wd = ((const float4*)gw)[lane];
  const float4 ws = ((const float4*)(gw + GCN_HIDDEN))[lane];
  float d1 = a.x * wd.x + a.y * wd.y + a.z * wd.z + a.w * wd.w;
  float d2 = a.x * ws.x + a.y * ws.y + a.z * ws.z + a.w * ws.w;
#pragma unroll
  for (int off = 16; off > 0; off >>= 1) {
    d1 += __shfl_xor(d1, off, 32);
    d2 += __shfl_xor(d2, off, 32);
  }
  if (lane == 0) {
    sdst[node] = d1;
    ssrc[node] = d2;
  }
}

// ---------------------------------------------------------------------------
// z = EPS * raw  (elementwise init before scatter-add)
// ---------------------------------------------------------------------------
__global__ __launch_bounds__(256) void init_z_kernel(
    const float* __restrict__ raw, float* __restrict__ z, int n4) {
  const int i = blockIdx.x * blockDim.x + threadIdx.x;
  if (i >= n4) return;
  float4 v = ((const float4*)raw)[i];
  v.x *= GCN_EPS; v.y *= GCN_EPS; v.z *= GCN_EPS; v.w *= GCN_EPS;
  ((float4*)z)[i] = v;
}

// ---------------------------------------------------------------------------
// Edge scatter: z[dst] += h[src] * tanh(sdst[dst]+ssrc[src]+gb) * d[dst]*d[src]
// One wave per edge; lane handles 4 channels with f32 atomics.
// ---------------------------------------------------------------------------
__global__ __launch_bounds__(256) void edge_scatter_kernel(
    const float* __restrict__ h, const int* __restrict__ src,
    const int* __restrict__ dst, const float* __restrict__ d,
    const float* __restrict__ sdst, const float* __restrict__ ssrc,
    const float* __restrict__ gate_b, int layer,
    float* __restrict__ z, int nedges) {
  const int e    = (int)((blockIdx.x * (unsigned)blockDim.x + threadIdx.x) >> 5);
  const int lane = threadIdx.x & 31;
  if (e >= nedges) return;

  const int s = src[e];
  const int t = dst[e];
  const float g    = tanhf(sdst[t] + ssrc[s] + gate_b[layer]);
  const float coef = g * d[t] * d[s];

  const float4 a = ((const float4*)(h + (size_t)s * GCN_HIDDEN))[lane];
  float* zp = z + (size_t)t * GCN_HIDDEN + lane * 4;
  atomicAdd(zp + 0, a.x * coef);
  atomicAdd(zp + 1, a.y * coef);
  atomicAdd(zp + 2, a.z * coef);
  atomicAdd(zp + 3, a.w * coef);
}

// ---------------------------------------------------------------------------
// In-place log_softmax over rows of 64. One wave per row, 2 cols per lane.
// ---------------------------------------------------------------------------
__global__ __launch_bounds__(256) void log_softmax_kernel(
    float* __restrict__ out, int n) {
  const int row  = (int)((blockIdx.x * (unsigned)blockDim.x + threadIdx.x) >> 5);
  const int lane = threadIdx.x & 31;
  if (row >= n) return;

  float* p = out + (size_t)row * GCN_OUT_DIM;
  float x0 = p[lane];
  float x1 = p[lane + 32];
  float m = fmaxf(x0, x1);
#pragma unroll
  for (int off = 16; off > 0; off >>= 1) m = fmaxf(m, __shfl_xor(m, off, 32));
  float s = __expf(x0 - m) + __expf(x1 - m);
#pragma unroll
  for (int off = 16; off > 0; off >>= 1) s += __shfl_xor(s, off, 32);
  const float ls = m + __logf(s);
  p[lane]      = x0 - ls;
  p[lane + 32] = x1 - ls;
}

// ---------------------------------------------------------------------------
extern "C" void kernel_launch(void* const* d_in, const int* in_sizes, int n_in,
                              void* d_out, int out_size, void* d_ws, size_t ws_size,
                              hipStream_t stream) {
  const float* h   = (const float*)d_in[0];  // [N,512]
  const int*   src = (const int*)d_in[1];    // [E]
  const int*   dst = (const int*)d_in[2];    // [E]
  const float* d   = (const float*)d_in[3];  // [N]
  const float* t1w = (const float*)d_in[4];  // [128,512]
  const float* t1b = (const float*)d_in[5];  // [128]
  const float* gw  = (const float*)d_in[6];  // [2,256]
  const float* gb  = (const float*)d_in[7];  // [2]
  const float* t2w = (const float*)d_in[8];  // [64,128]
  const float* t2b = (const float*)d_in[9];  // [64]
  float* out = (float*)d_out;                // [N,64]

  const int n_nodes = in_sizes[3];
  const int n_edges = in_sizes[1];

  float* raw  = (float*)d_ws;
  float* hb1  = raw + (size_t)n_nodes * GCN_HIDDEN;
  float* hb2  = hb1 + (size_t)n_nodes * GCN_HIDDEN;
  float* sdst = hb2 + (size_t)n_nodes * GCN_HIDDEN;
  float* ssrc = sdst + n_nodes;

  const int mtiles = (n_nodes + 15) / 16;  // 50000 is an exact multiple of 16

  // h1 = relu(h @ t1_w^T + t1_b)  -> raw
  gemm1_relu_kernel<<<mtiles, 256, 0, stream>>>(h, t1w, t1b, raw);

  const float* hcur = raw;
  for (int layer = 0; layer < 2; ++layer) {
    float* znext = (layer == 0) ? hb1 : hb2;
    gate_scores_kernel<<<(n_nodes + 7) / 8, 256, 0, stream>>>(
        hcur, gw + layer * 2 * GCN_HIDDEN, sdst, ssrc, n_nodes);
    const int n4 = n_nodes * GCN_HIDDEN / 4;
    init_z_kernel<<<(n4 + 255) / 256, 256, 0, stream>>>(raw, znext, n4);
    edge_scatter_kernel<<<(n_edges + 7) / 8, 256, 0, stream>>>(
        hcur, src, dst, d, sdst, ssrc, gb, layer, znext, n_edges);
    hcur = znext;
  }

  // logits -> d_out, then in-place log_softmax
  gemm2_kernel<<<mtiles, 128, 0, stream>>>(hcur, t2w, t2b, out);
  log_softmax_kernel<<<(n_nodes + 7) / 8, 256, 0, stream>>>(out, n_nodes);
}